// TransformerAttentionModule_59390807769625
// MI455X (gfx1250) — compile-verified
//
#include <hip/hip_runtime.h>
#include <hip/hip_bf16.h>

typedef __attribute__((ext_vector_type(2))) float v2f;
typedef __attribute__((ext_vector_type(8))) float v8f;

#define HD   8     // heads
#define DD   128   // model dim
#define DHD  16    // head dim

// ---------------------------------------------------------------------------
// C (nrows x ncols) = A (nrows x 128) @ W^T (W is ncols x 128, row-major) + bias
// One block = 256 threads = 8 waves. Block covers 16 rows x 128 cols
// (grid.y selects which 128-col group). Uses V_WMMA_F32_16X16X4_F32.
// A tile staged in LDS (padded stride 132 -> conflict-free frag reads).
// ---------------------------------------------------------------------------
__global__ __launch_bounds__(256) void gemm_nt_wmma_f32(
    const float* __restrict__ A, const float* __restrict__ W,
    const float* __restrict__ bias, float* __restrict__ C, int ncols)
{
    __shared__ float lds[16 * 132];
    const int tid     = threadIdx.x;
    const int rowBase = blockIdx.x * 16;

    // Cooperative load of the 16x128 A tile (8 floats per thread, coalesced).
    {
        const int r = tid >> 4;           // 0..15
        const int c = (tid & 15) * 8;     // 0..120
        const float4* g = (const float4*)(A + (size_t)(rowBase + r) * DD + c);
        float4 v0 = g[0];
        float4 v1 = g[1];
        float* s = &lds[r * 132 + c];
        s[0] = v0.x; s[1] = v0.y; s[2] = v0.z; s[3] = v0.w;
        s[4] = v1.x; s[5] = v1.y; s[6] = v1.z; s[7] = v1.w;
    }
    __syncthreads();

    const int wave = tid >> 5;
    const int lane = tid & 31;
    const int half = lane >> 4;           // 0: lanes 0-15, 1: lanes 16-31
    const int m    = lane & 15;
    const int col  = blockIdx.y * 128 + wave * 16 + m;   // output feature = W row

    const float* wrow = W + (size_t)col * DD;
    const float* arow = &lds[m * 132];

    v8f acc = {};
#pragma unroll
    for (int k = 0; k < DD; k += 4) {
        const int kk = k + 2 * half;
        // A 16x4 frag: lane(m), VGPR j = A[m][k + 2*half + j]
        v2f a = { arow[kk], arow[kk + 1] };
        // B 4x16 frag: lane(n=m), VGPR j = B[k + 2*half + j][n] = W[col][kk+j]
        v2f b = { wrow[kk], wrow[kk + 1] };
        acc = __builtin_amdgcn_wmma_f32_16x16x4_f32(
            /*neg_a=*/false, a, /*neg_b=*/false, b,
            /*c_mod=*/(short)0, acc, /*reuse_a=*/false, /*reuse_b=*/false);
    }

    const float bv = bias[col];
#pragma unroll
    for (int r = 0; r < 8; ++r) {
        const int row = rowBase + r + 8 * half;   // C/D layout: M = r + 8*half
        C[(size_t)row * ncols + col] = acc[r] + bv;
    }
}

// ---------------------------------------------------------------------------
// Workspace init: smax = encode(-inf), denom = 0, agg = 0
// encode(f): order-preserving float->uint. encode(-inf) = 0x007FFFFF.
// ---------------------------------------------------------------------------
__global__ __launch_bounds__(256) void init_kernel(
    unsigned* __restrict__ smax, float* __restrict__ denom,
    float* __restrict__ agg, int n8, int n128)
{
    const int i = blockIdx.x * 256 + threadIdx.x;
    if (i < n8) { smax[i] = 0x007FFFFFu; denom[i] = 0.0f; }
    if (i < n128) agg[i] = 0.0f;
}

__device__ __forceinline__ unsigned enc_f32(float f) {
    int b = __float_as_int(f);
    return (unsigned)(b ^ ((b >> 31) | 0x80000000));
}
__device__ __forceinline__ float dec_f32(unsigned u) {
    int b = (u & 0x80000000u) ? (int)(u ^ 0x80000000u) : (int)~u;
    return __int_as_float(b);
}

// ---------------------------------------------------------------------------
// scores[e,h] = (k[src,h] . q[dst,h]) / sqrt(DH); atomic segment-max into smax
// qkv row layout per head h: [q(16) | k(16) | v(16)] at offset h*48
// ---------------------------------------------------------------------------
__global__ __launch_bounds__(256) void score_kernel(
    const float* __restrict__ qkv, const int* __restrict__ src,
    const int* __restrict__ dst, float* __restrict__ scores,
    unsigned* __restrict__ smax, int eh)
{
    const int i = blockIdx.x * 256 + threadIdx.x;
    if (i >= eh) return;
    const int e = i >> 3, h = i & 7;
    const int s = src[e], d = dst[e];
    const float4* kp = (const float4*)(qkv + (size_t)s * 384 + h * 48 + DHD);
    const float4* qp = (const float4*)(qkv + (size_t)d * 384 + h * 48);
    float acc = 0.0f;
#pragma unroll
    for (int j = 0; j < 4; ++j) {
        float4 kv = kp[j], qv = qp[j];
        acc += kv.x * qv.x + kv.y * qv.y + kv.z * qv.z + kv.w * qv.w;
    }
    const float sc = acc * 0.25f;   // 1/sqrt(16)
    scores[i] = sc;
    atomicMax(&smax[(size_t)d * HD + h], enc_f32(sc));
}

// ---------------------------------------------------------------------------
// ex = exp(score - smax[dst]); atomic segment-sum into denom; ex overwrites score
// ---------------------------------------------------------------------------
__global__ __launch_bounds__(256) void exp_kernel(
    const int* __restrict__ dst, float* __restrict__ scores,
    const unsigned* __restrict__ smax, float* __restrict__ denom, int eh)
{
    const int i = blockIdx.x * 256 + threadIdx.x;
    if (i >= eh) return;
    const int e = i >> 3, h = i & 7;
    const int d = dst[e];
    float mx = dec_f32(smax[(size_t)d * HD + h]);
    if (!isfinite(mx)) mx = 0.0f;
    const float ex = expf(scores[i] - mx);
    scores[i] = ex;
    atomicAdd(&denom[(size_t)d * HD + h], ex);
}

// ---------------------------------------------------------------------------
// agg[dst, d] += (ex / max(denom,1e-30)) * v[src, d]; one thread per (edge,feat)
// ---------------------------------------------------------------------------
__global__ __launch_bounds__(256) void agg_kernel(
    const float* __restrict__ qkv, const int* __restrict__ src,
    const int* __restrict__ dst, const float* __restrict__ scores,
    const float* __restrict__ denom, float* __restrict__ agg, int total)
{
    const int i = blockIdx.x * 256 + threadIdx.x;
    if (i >= total) return;
    const int e = i >> 7;
    const int dc = i & 127;          // feature index 0..127
    const int h = dc >> 4;
    const int d = dst[e];
    const float ex  = scores[(size_t)e * HD + h];
    const float den = denom[(size_t)d * HD + h];
    const float p   = ex / fmaxf(den, 1e-30f);
    const float v   = qkv[(size_t)src[e] * 384 + h * 48 + 32 + (dc & 15)];
    atomicAdd(&agg[(size_t)d * DD + dc], p * v);
}

// ---------------------------------------------------------------------------
extern "C" void kernel_launch(void* const* d_in, const int* in_sizes, int n_in,
                              void* d_out, int out_size, void* d_ws, size_t ws_size,
                              hipStream_t stream) {
    const float* x     = (const float*)d_in[0];
    const int*   src   = (const int*)  d_in[1];
    const int*   dst   = (const int*)  d_in[2];
    const float* w_qkv = (const float*)d_in[3];
    const float* b_qkv = (const float*)d_in[4];
    const float* w_out = (const float*)d_in[5];
    const float* b_out = (const float*)d_in[6];
    float* out = (float*)d_out;

    const int n = in_sizes[0] / DD;   // 50000 (multiple of 16)
    const int e = in_sizes[1];        // 800000

    // Workspace carve-out (256B aligned): qkv 76.8MB, scores 25.6MB,
    // smax/denom 1.6MB each, agg 25.6MB  (~131MB total)
    char* p = (char*)d_ws;
    auto take = [&](size_t bytes) {
        char* q = p;
        p += (bytes + 255) & ~(size_t)255;
        return q;
    };
    float*    qkv    = (float*)   take((size_t)n * 384 * sizeof(float));
    float*    scores = (float*)   take((size_t)e * HD * sizeof(float));
    unsigned* smax   = (unsigned*)take((size_t)n * HD * sizeof(unsigned));
    float*    denom  = (float*)   take((size_t)n * HD * sizeof(float));
    float*    agg    = (float*)   take((size_t)n * DD * sizeof(float));

    const int n8 = n * HD, n128 = n * DD;
    hipLaunchKernelGGL(init_kernel, dim3((n128 + 255) / 256), dim3(256), 0, stream,
                       smax, denom, agg, n8, n128);

    // qkv = x @ w_qkv^T + b_qkv   (N x 384)
    hipLaunchKernelGGL(gemm_nt_wmma_f32, dim3(n / 16, 3), dim3(256), 0, stream,
                       x, w_qkv, b_qkv, qkv, 384);

    const int eh = e * HD;
    hipLaunchKernelGGL(score_kernel, dim3((eh + 255) / 256), dim3(256), 0, stream,
                       qkv, src, dst, scores, smax, eh);
    hipLaunchKernelGGL(exp_kernel, dim3((eh + 255) / 256), dim3(256), 0, stream,
                       dst, scores, smax, denom, eh);

    const int tot = e * DD;
    hipLaunchKernelGGL(agg_kernel, dim3((tot + 255) / 256), dim3(256), 0, stream,
                       qkv, src, dst, scores, denom, agg, tot);

    // out = agg @ w_out^T + b_out   (N x 128)
    hipLaunchKernelGGL(gemm_nt_wmma_f32, dim3(n / 16, 1), dim3(256), 0, stream,
                       agg, w_out, b_out, out, DD);
}